// GraphConvolution_3453153706335
// MI455X (gfx1250) — compile-verified
//
#include <hip/hip_runtime.h>

typedef __attribute__((ext_vector_type(2))) float v2f;
typedef __attribute__((ext_vector_type(8))) float v8f;

#define D 128              // D_IN == D_OUT == 128
#define LDA 132            // padded LDS row stride (132 % 64 = 4 -> conflict-free column reads)

// ---------------------------------------------------------------------------
// Kernel 1: support = X @ W   via V_WMMA_F32_16X16X4_F32
// grid.x = M/16 blocks, 256 threads (8 waves). Wave w computes the 16x16 tile
// at columns [16w, 16w+16) for this block's 16 rows.
// ---------------------------------------------------------------------------
__global__ void __launch_bounds__(256)
gcn_gemm_xw_wmma(const float* __restrict__ X, const float* __restrict__ W,
                 float* __restrict__ S, int M) {
    __shared__ float As[16 * LDA];

    const int tid = threadIdx.x;
    const int m0  = blockIdx.x * 16;

    // Stage the 16x128 A tile into LDS (coalesced, row clamped for tail safety).
    #pragma unroll
    for (int i = 0; i < 8; ++i) {
        int g = tid + i * 256;          // 0..2047
        int r = g >> 7;                 // row 0..15
        int c = g & 127;                // col 0..127
        int rg = m0 + r; rg = rg < M ? rg : (M - 1);   // clamp, no EXEC divergence
        As[r * LDA + c] = X[(size_t)rg * D + c];
    }
    __syncthreads();

    const int wave = tid >> 5;
    const int lane = tid & 31;
    const int half = lane >> 4;         // 0: K pair {0,1}, 1: K pair {2,3}
    const int mn   = lane & 15;         // M index for A, N index for B/C/D
    const int nb   = wave * 16;         // this wave's output-column base

    // Preload this wave's 128x16 slice of W into registers (W is 64KB, L2-hot).
    float breg[64];
    #pragma unroll
    for (int t = 0; t < 32; ++t) {
        int k0 = 4 * t + 2 * half;
        breg[2 * t + 0] = W[(size_t)(k0 + 0) * D + nb + mn];
        breg[2 * t + 1] = W[(size_t)(k0 + 1) * D + nb + mn];
    }

    v8f acc = {};
    #pragma unroll
    for (int t = 0; t < 32; ++t) {
        int k0 = 4 * t + 2 * half;
        v2f a, b;
        a.x = As[mn * LDA + k0];
        a.y = As[mn * LDA + k0 + 1];
        b.x = breg[2 * t + 0];
        b.y = breg[2 * t + 1];
        // D = A(16x4) * B(4x16) + C ; 8 args: neg_a, A, neg_b, B, c_mod, C, reuse_a, reuse_b
        acc = __builtin_amdgcn_wmma_f32_16x16x4_f32(false, a, false, b,
                                                    (short)0, acc, false, false);
    }

    // C/D layout: VGPR r holds M = r (lanes 0-15) / r+8 (lanes 16-31), N = lane%16.
    #pragma unroll
    for (int r = 0; r < 8; ++r) {
        int mr = m0 + r + 8 * half;
        if (mr < M) S[(size_t)mr * D + nb + mn] = acc[r];
    }
}

// ---------------------------------------------------------------------------
// Kernel 2: out[n][d] = bias[d]  (also provides required zero-init + bias term)
// ---------------------------------------------------------------------------
__global__ void __launch_bounds__(256)
gcn_bias_init(float* __restrict__ out, const float* __restrict__ bias, int total) {
    int idx = blockIdx.x * 256 + threadIdx.x;
    if (idx < total) out[idx] = bias[idx & (D - 1)];
}

// ---------------------------------------------------------------------------
// Kernel 3: out[dst] += edge_weight * support[src]   (one wave32 per edge,
// each lane owns a float4 => 32*4 = 128 channels). support & out are L2-resident
// (51.2MB each < 192MB L2), so gathers and fp32 atomics stay on-chip.
// ---------------------------------------------------------------------------
__global__ void __launch_bounds__(256)
gcn_edge_scatter(const float* __restrict__ support,
                 const int* __restrict__ edge_src,
                 const int* __restrict__ edge_dst,
                 const float* __restrict__ edge_w,
                 float* __restrict__ out, int E) {
    int e = (blockIdx.x * 256 + threadIdx.x) >> 5;   // one wave per edge
    if (e >= E) return;
    int lane = threadIdx.x & 31;

    int   s  = edge_src[e];
    int   d  = edge_dst[e];
    float ew = edge_w[e];

    const float4 v = reinterpret_cast<const float4*>(support + (size_t)s * D)[lane];
    float* o = out + (size_t)d * D + lane * 4;

    unsafeAtomicAdd(o + 0, v.x * ew);
    unsafeAtomicAdd(o + 1, v.y * ew);
    unsafeAtomicAdd(o + 2, v.z * ew);
    unsafeAtomicAdd(o + 3, v.w * ew);
}

// ---------------------------------------------------------------------------
extern "C" void kernel_launch(void* const* d_in, const int* in_sizes, int n_in,
                              void* d_out, int out_size, void* d_ws, size_t ws_size,
                              hipStream_t stream) {
    const float* x     = (const float*)d_in[0];
    const int*   esrc  = (const int*)  d_in[1];
    const int*   edst  = (const int*)  d_in[2];
    const float* ew    = (const float*)d_in[3];
    const float* w     = (const float*)d_in[4];
    const float* bias  = (const float*)d_in[5];
    float*       out   = (float*)d_out;

    const int M = in_sizes[0] / D;     // 100000 nodes
    const int E = in_sizes[1];         // 1600000 edges

    float* support = (float*)d_ws;     // M * D fp32 = 51.2 MB scratch

    // 1) support = X @ W  (WMMA fp32)
    gcn_gemm_xw_wmma<<<(M + 15) / 16, 256, 0, stream>>>(x, w, support, M);

    // 2) out = bias (broadcast)
    gcn_bias_init<<<(M * D + 255) / 256, 256, 0, stream>>>(out, bias, M * D);

    // 3) out[dst] += ew * support[src]
    gcn_edge_scatter<<<(E + 7) / 8, 256, 0, stream>>>(support, esrc, edst, ew, out, E);
}